// MY_TGCN__30709016166899
// MI455X (gfx1250) — compile-verified
//
#include <hip/hip_runtime.h>
#include <hip/hip_bf16.h>
#include <cstdint>

// ---------------------------------------------------------------------------
// T-GCN cell for MI455X (gfx1250, wave32).
// Dense GEMMs use V_WMMA_F32_16X16X4_F32 (full fp32 — the workload is
// scatter-bound, so low-precision WMMA buys nothing; fp32 matches reference).
// ---------------------------------------------------------------------------

typedef __attribute__((ext_vector_type(2))) float v2f;
typedef __attribute__((ext_vector_type(8))) float v8f;

__device__ __forceinline__ v8f wmma_f32x4(v2f a, v2f b, v8f c) {
  // D(16x16,f32) += A(16x4,f32) * B(4x16,f32)
  return __builtin_amdgcn_wmma_f32_16x16x4_f32(
      /*neg_a=*/false, a, /*neg_b=*/false, b,
      /*c_mod=*/(short)0, c, /*reuse_a=*/false, /*reuse_b=*/false);
}

__device__ __forceinline__ float sigmoidf_(float t) {
  return 1.0f / (1.0f + __expf(-t));
}

// ------------------------- normalization kernels ---------------------------

__global__ void k_deg_init(float* __restrict__ deg, int N) {
  int i = blockIdx.x * blockDim.x + threadIdx.x;
  if (i < N) deg[i] = 1.0f;               // self-loop weight
}

__global__ void k_deg_accum(const int* __restrict__ dst,
                            const float* __restrict__ w,
                            float* __restrict__ deg, int E) {
  int e = blockIdx.x * blockDim.x + threadIdx.x;
  if (e < E) atomicAdd(&deg[dst[e]], w[e]);
}

__global__ void k_rsqrt(float* __restrict__ dinv, int N) {
  int i = blockIdx.x * blockDim.x + threadIdx.x;
  if (i < N) {
    float d = dinv[i];
    dinv[i] = (d > 0.0f) ? rsqrtf(d) : 0.0f;
  }
}

__global__ void k_norm(const int* __restrict__ src, const int* __restrict__ dst,
                       const float* __restrict__ w, const float* __restrict__ dinv,
                       float* __restrict__ norm, int E) {
  int e = blockIdx.x * blockDim.x + threadIdx.x;
  if (e < E) norm[e] = dinv[src[e]] * w[e] * dinv[dst[e]];
}

// ------------------- XW = x @ [Wz | Wr | Wh]  (N x 192) --------------------
// One wave per 16-row tile. A fragments preloaded (16x128 fp32 = 64 VGPRs).

__global__ void k_gemm_xw(const float* __restrict__ x,
                          const float* __restrict__ Wz,
                          const float* __restrict__ Wr,
                          const float* __restrict__ Wh,
                          float* __restrict__ XW, int N) {
  int wave = threadIdx.x >> 5;
  int lane = threadIdx.x & 31;
  int tile = blockIdx.x * (blockDim.x >> 5) + wave;
  int numTiles = (N + 15) >> 4;
  if (tile >= numTiles) return;           // wave-uniform: EXEC stays all-1s

  int m0 = tile << 4;
  int m  = lane & 15;
  int kh = lane >> 4;                     // selects K pair {2kh, 2kh+1}
  int n  = lane & 15;
  int row = m0 + m; if (row > N - 1) row = N - 1;
  const float* xrow = x + (size_t)row * 128;

  v2f afrag[32];
#pragma unroll
  for (int kk = 0; kk < 32; ++kk) {
    int kb = 4 * kk + 2 * kh;
    afrag[kk].x = xrow[kb];
    afrag[kk].y = xrow[kb + 1];
  }

  for (int c = 0; c < 12; ++c) {
    const float* Wm = (c < 4) ? Wz : (c < 8) ? Wr : Wh;
    int ncol = (c & 3) * 16 + n;
    v8f acc = {};
#pragma unroll
    for (int kk = 0; kk < 32; ++kk) {
      int kb = 4 * kk + 2 * kh;
      v2f b;
      b.x = Wm[(size_t)kb * 64 + ncol];
      b.y = Wm[(size_t)(kb + 1) * 64 + ncol];
      acc = wmma_f32x4(afrag[kk], b, acc);
    }
    int colg = c * 16 + n;                // [z|r|h] blocks contiguous
#pragma unroll
    for (int v = 0; v < 8; ++v) {
      int rm = m0 + v + 8 * kh;           // C layout: row = v + 8*(lane>=16)
      if (rm < N) XW[(size_t)rm * 192 + colg] = acc[v];
    }
  }
}

// ----------------- agg init: self-loop term + bias -------------------------

__global__ void k_agg_init(const float* __restrict__ XW,
                           const float* __restrict__ dinv,
                           const float* __restrict__ bz,
                           const float* __restrict__ br,
                           const float* __restrict__ bh,
                           float* __restrict__ agg, int N) {
  int i = blockIdx.x * blockDim.x + threadIdx.x;
  int total = N * 192;
  if (i >= total) return;
  int node = i / 192;
  int j = i - node * 192;
  float b = (j < 64) ? bz[j] : (j < 128) ? br[j - 64] : bh[j - 128];
  float di = dinv[node];
  agg[i] = di * di * XW[i] + b;
}

// ---------------- edge scatter: agg[d] += norm * XW[s] ---------------------
// 48 float4 chunks per edge; consecutive lanes walk one edge's feature row.

__global__ void k_edge_scatter(const int* __restrict__ src,
                               const int* __restrict__ dst,
                               const float* __restrict__ norm,
                               const float* __restrict__ XW,
                               float* __restrict__ agg, int E) {
  int tid = blockIdx.x * blockDim.x + threadIdx.x;
  int total = E * 48;
  if (tid >= total) return;
  int e = tid / 48;
  int f = (tid - e * 48) * 4;
  float w = norm[e];
  int s = src[e], d = dst[e];
  const float4 v = *(const float4*)(XW + (size_t)s * 192 + f);
  float* a = agg + (size_t)d * 192 + f;
  atomicAdd(a + 0, w * v.x);
  atomicAdd(a + 1, w * v.y);
  atomicAdd(a + 2, w * v.z);
  atomicAdd(a + 3, w * v.w);
}

// --------------------------- fused gate kernel -----------------------------
// Per 16-row tile (one wave):
//   R  = sigmoid([agg_r | H] @ Lr + lbr);  HR = H*R  -> LDS (layout transpose)
//   Z  = sigmoid([agg_z | H] @ Lz + lbz)
//   Ht = tanh   ([agg_h | HR] @ Lh + lbh)   (HR K-half read back from LDS)
//   Hn = Z*H + (1-Z)*Ht

__global__ void k_gates(const float* __restrict__ agg,
                        const float* __restrict__ H,
                        const float* __restrict__ Lz, const float* __restrict__ lbz,
                        const float* __restrict__ Lr, const float* __restrict__ lbr,
                        const float* __restrict__ Lh, const float* __restrict__ lbh,
                        float* __restrict__ Hnew, int N) {
  __shared__ float ldsHR[8][16 * 64];     // 32 KB: per-wave transpose scratch

  int wave = threadIdx.x >> 5;
  int lane = threadIdx.x & 31;
  int tile = blockIdx.x * (blockDim.x >> 5) + wave;
  int numTiles = (N + 15) >> 4;
  if (tile >= numTiles) return;           // wave-uniform

  int m0 = tile << 4;
  int m  = lane & 15;
  int kh = lane >> 4;
  int n  = lane & 15;
  int row = m0 + m; if (row > N - 1) row = N - 1;
  const float* aggrow = agg + (size_t)row * 192;
  const float* hrow   = H   + (size_t)row * 64;
  float* myHR = ldsHR[wave];

  // ---- pass 1: R, HR = H * R ----
  for (int c = 0; c < 4; ++c) {
    int ncol = c * 16 + n;
    v8f acc = {};
#pragma unroll
    for (int kk = 0; kk < 16; ++kk) {     // K 0..63 from agg_r block
      int kb = 4 * kk + 2 * kh;
      v2f a; a.x = aggrow[64 + kb]; a.y = aggrow[64 + kb + 1];
      v2f b; b.x = Lr[(size_t)kb * 64 + ncol]; b.y = Lr[(size_t)(kb + 1) * 64 + ncol];
      acc = wmma_f32x4(a, b, acc);
    }
#pragma unroll
    for (int kk = 0; kk < 16; ++kk) {     // K 64..127 from H
      int kb = 4 * kk + 2 * kh;
      v2f a; a.x = hrow[kb]; a.y = hrow[kb + 1];
      v2f b; b.x = Lr[(size_t)(64 + kb) * 64 + ncol]; b.y = Lr[(size_t)(65 + kb) * 64 + ncol];
      acc = wmma_f32x4(a, b, acc);
    }
    float bias = lbr[ncol];
#pragma unroll
    for (int v = 0; v < 8; ++v) {
      int rm = v + 8 * kh;                // local row of this tile
      int gr = m0 + rm; if (gr > N - 1) gr = N - 1;
      float r = sigmoidf_(acc[v] + bias);
      float h = H[(size_t)gr * 64 + ncol];
      myHR[rm * 64 + ncol] = h * r;
    }
  }

  // CDNA5 split counter: drain LDS stores before cross-lane LDS reads.
  asm volatile("s_wait_dscnt 0" ::: "memory");

  // ---- pass 2: Z, H~, H_new ----
  for (int c = 0; c < 4; ++c) {
    int ncol = c * 16 + n;
    v8f accZ = {};
    v8f accT = {};
#pragma unroll
    for (int kk = 0; kk < 16; ++kk) {     // K 0..63: agg_z / agg_h blocks
      int kb = 4 * kk + 2 * kh;
      v2f az; az.x = aggrow[kb];        az.y = aggrow[kb + 1];
      v2f ah; ah.x = aggrow[128 + kb];  ah.y = aggrow[128 + kb + 1];
      v2f bz2; bz2.x = Lz[(size_t)kb * 64 + ncol]; bz2.y = Lz[(size_t)(kb + 1) * 64 + ncol];
      v2f bh2; bh2.x = Lh[(size_t)kb * 64 + ncol]; bh2.y = Lh[(size_t)(kb + 1) * 64 + ncol];
      accZ = wmma_f32x4(az, bz2, accZ);
      accT = wmma_f32x4(ah, bh2, accT);
    }
#pragma unroll
    for (int kk = 0; kk < 16; ++kk) {     // K 64..127: H (for Z), HR (for H~)
      int kb = 4 * kk + 2 * kh;
      v2f az; az.x = hrow[kb]; az.y = hrow[kb + 1];
      v2f ah = *(const v2f*)&myHR[m * 64 + kb];   // LDS, 8B aligned (kb even)
      v2f bz2; bz2.x = Lz[(size_t)(64 + kb) * 64 + ncol]; bz2.y = Lz[(size_t)(65 + kb) * 64 + ncol];
      v2f bh2; bh2.x = Lh[(size_t)(64 + kb) * 64 + ncol]; bh2.y = Lh[(size_t)(65 + kb) * 64 + ncol];
      accZ = wmma_f32x4(az, bz2, accZ);
      accT = wmma_f32x4(ah, bh2, accT);
    }
    float biasZ = lbz[ncol];
    float biasH = lbh[ncol];
#pragma unroll
    for (int v = 0; v < 8; ++v) {
      int gr = m0 + v + 8 * kh;
      if (gr < N) {
        float z  = sigmoidf_(accZ[v] + biasZ);
        float ht = tanhf(accT[v] + biasH);
        float h  = H[(size_t)gr * 64 + ncol];
        Hnew[(size_t)gr * 64 + ncol] = z * h + (1.0f - z) * ht;
      }
    }
  }
}

// ------------------------ y = relu(H_new) @ Wo + bo ------------------------

__global__ void k_out_proj(const float* __restrict__ Hnew,
                           const float* __restrict__ Wo,
                           const float* __restrict__ bo,
                           float* __restrict__ y, int N) {
  int i = blockIdx.x * blockDim.x + threadIdx.x;
  if (i >= N) return;
  const float* h = Hnew + (size_t)i * 64;
  float acc = bo[0];
#pragma unroll
  for (int j = 0; j < 64; ++j) acc += fmaxf(h[j], 0.0f) * Wo[j];
  y[i] = acc;
}

// ------------------------------- launcher ----------------------------------

extern "C" void kernel_launch(void* const* d_in, const int* in_sizes, int n_in,
                              void* d_out, int out_size, void* d_ws, size_t ws_size,
                              hipStream_t stream) {
  const float* x   = (const float*)d_in[0];
  const int*   ei  = (const int*)d_in[1];
  const float* ew  = (const float*)d_in[2];
  const float* H   = (const float*)d_in[3];
  // d_in[4] = c (unused by reference)
  const float* Wz  = (const float*)d_in[5];
  const float* bz  = (const float*)d_in[6];
  const float* Wr  = (const float*)d_in[7];
  const float* br  = (const float*)d_in[8];
  const float* Wh  = (const float*)d_in[9];
  const float* bh  = (const float*)d_in[10];
  const float* Lz  = (const float*)d_in[11];
  const float* lbz = (const float*)d_in[12];
  const float* Lr  = (const float*)d_in[13];
  const float* lbr = (const float*)d_in[14];
  const float* Lh  = (const float*)d_in[15];
  const float* lbh = (const float*)d_in[16];
  const float* Wo  = (const float*)d_in[17];
  const float* bo  = (const float*)d_in[18];

  const int N = in_sizes[3] / 64;
  const int E = in_sizes[2];
  const int* src = ei;
  const int* dst = ei + E;

  // workspace layout (floats): dinv[N] | norm[E] | XW[N*192] | agg[N*192]
  float* ws   = (float*)d_ws;
  float* dinv = ws;
  float* norm = dinv + N;
  float* XW   = norm + E;
  float* agg  = XW + (size_t)N * 192;

  float* yout = (float*)d_out;            // y: N, then H_new: N*64
  float* Hnew = yout + N;

  const int B = 256;
  k_deg_init  <<<(N + B - 1) / B, B, 0, stream>>>(dinv, N);
  k_deg_accum <<<(E + B - 1) / B, B, 0, stream>>>(dst, ew, dinv, E);
  k_rsqrt     <<<(N + B - 1) / B, B, 0, stream>>>(dinv, N);
  k_norm      <<<(E + B - 1) / B, B, 0, stream>>>(src, dst, ew, dinv, norm, E);

  const int numTiles = (N + 15) / 16;     // 3125 for N=50000
  const int gw = (numTiles + 7) / 8;      // 8 waves (16-row tiles) per block
  k_gemm_xw   <<<gw, B, 0, stream>>>(x, Wz, Wr, Wh, XW, N);

  const int aggTot = N * 192;
  k_agg_init  <<<(aggTot + B - 1) / B, B, 0, stream>>>(XW, dinv, bz, br, bh, agg, N);

  const int scatTot = E * 48;             // 48 float4 chunks per edge
  k_edge_scatter<<<(scatTot + B - 1) / B, B, 0, stream>>>(src, dst, norm, XW, agg, E);

  k_gates     <<<gw, B, 0, stream>>>(agg, H, Lz, lbz, Lr, lbr, Lh, lbh, Hnew, N);
  k_out_proj  <<<(N + B - 1) / B, B, 0, stream>>>(Hnew, Wo, bo, yout, N);
}